// GINConvNet_910533067566
// MI455X (gfx1250) — compile-verified
//
#include <hip/hip_runtime.h>
#include <hip/hip_bf16.h>

typedef float v2f __attribute__((ext_vector_type(2)));
typedef float v8f __attribute__((ext_vector_type(8)));

#define NN      100000
#define EE      640000
#define DIMC    128
#define NGRAPH  512
#define BN_EPS  1e-5f

// Async global->LDS staging if this toolchain exposes the gfx1250 builtins.
#if defined(__has_builtin)
#if __has_builtin(__builtin_amdgcn_global_load_async_to_lds_b128) && \
    __has_builtin(__builtin_amdgcn_s_wait_asynccnt)
#define USE_ASYNC_LDS 1
#endif
#endif
#ifndef USE_ASYNC_LDS
#define USE_ASYNC_LDS 0
#endif

#if USE_ASYNC_LDS
typedef int b128_t __attribute__((vector_size(16)));
typedef __attribute__((address_space(1))) b128_t* gas_b128;
typedef __attribute__((address_space(3))) b128_t* las_b128;
#endif

// ---------------------------------------------------------------------------
__global__ void copy_kernel(float* __restrict__ dst, const float* __restrict__ src, size_t n) {
    size_t i = (size_t)blockIdx.x * blockDim.x + threadIdx.x;
    if (i < n) dst[i] = src[i];
}

__global__ void zero_kernel(float* __restrict__ p, size_t n) {
    size_t i = (size_t)blockIdx.x * blockDim.x + threadIdx.x;
    if (i < n) p[i] = 0.0f;
}

// ---------------------------------------------------------------------------
// GIN aggregation: agg[dst[e]] += x[src[e]]  (agg pre-initialized to x)
__global__ void edge_scatter_kernel(float* __restrict__ agg, const float* __restrict__ x,
                                    const int* __restrict__ src, const int* __restrict__ dst,
                                    int nedges) {
    int gtid = blockIdx.x * blockDim.x + threadIdx.x;
    int e    = gtid >> 5;
    int lane = gtid & 31;
    if (e >= nedges) return;
    int s = src[e];
    int d = dst[e];
    const float4* xs = (const float4*)(x + (size_t)s * DIMC);
    float4 v = xs[lane];
    float* ap = agg + (size_t)d * DIMC + lane * 4;
    atomicAdd(ap + 0, v.x);
    atomicAdd(ap + 1, v.y);
    atomicAdd(ap + 2, v.z);
    atomicAdd(ap + 3, v.w);
}

// ---------------------------------------------------------------------------
// LDS-staged WMMA fp32 GEMM, compile-time K=N=128.
//   C[M,128] = act(A[M,128] @ W[128,128] + bias)
// W is staged TRANSPOSED in LDS (Wlds[n*WS+k]) so both A and B fragments are
// contiguous ds_load_b64, conflict-free with the +4 dword padded stride.
// Block = 256 thr = 8 waves; wave w owns output cols [16w,16w+16); grid-stride
// over 16-row tiles so W is loaded once per block.
template<int KD, int ND>
__global__ void __launch_bounds__(256)
gemm_conv_kernel(const float* __restrict__ A, const float* __restrict__ W,
                 const float* __restrict__ bias, float* __restrict__ C,
                 int M, int relu) {
    constexpr int WS = KD + 4;                 // padded k-stride (mult of 4 -> b128 ok)
    constexpr int AS = KD + 4;
    __shared__ float Wlds[ND * WS];            // transposed weights
    __shared__ float Alds[16 * AS];            // current 16-row A tile
    __shared__ float blds[ND];

    const int tid  = threadIdx.x;
    const int lane = tid & 31;
    const int wave = tid >> 5;
    const int half = lane >> 4;                // 0 | 1
    const int lm   = lane & 15;

    // ---- stage W (transposed) + bias, once per block ----
    for (int f = tid; f < (KD * ND) / 4; f += 256) {
        int k  = f >> 5;                       // ND/4 == 32 float4 per row
        int n4 = f & 31;
        float4 v = ((const float4*)(W + (size_t)k * ND))[n4];
        int n = n4 * 4;
        Wlds[(n + 0) * WS + k] = v.x;
        Wlds[(n + 1) * WS + k] = v.y;
        Wlds[(n + 2) * WS + k] = v.z;
        Wlds[(n + 3) * WS + k] = v.w;
    }
    if (tid < ND) blds[tid] = bias[tid];

    const int n      = (wave & 7) * 16 + lm;   // this wave's output column
    const int mtiles = M / 16;

    for (int t = blockIdx.x; t < mtiles; t += gridDim.x) {
        __syncthreads();                       // W ready / previous tile consumed
        // ---- stage A tile: 16 x 128 floats = 512 float4, 2 per thread ----
        #pragma unroll
        for (int rep = 0; rep < 2; ++rep) {
            int f   = tid + rep * 256;
            int row = f >> 5;                  // 32 float4 per row
            int c4  = f & 31;
            const float* gp = A + ((size_t)t * 16 + row) * KD + c4 * 4;
            float*       lp = &Alds[row * AS + c4 * 4];
#if USE_ASYNC_LDS
            __builtin_amdgcn_global_load_async_to_lds_b128(
                (gas_b128)(void*)gp, (las_b128)(void*)lp, 0, 0);
#else
            *(float4*)lp = *(const float4*)gp;
#endif
        }
#if USE_ASYNC_LDS
        __builtin_amdgcn_s_wait_asynccnt(0);
#endif
        __syncthreads();

        // ---- 16x16x128 accumulation: 32 chained v_wmma_f32_16x16x4_f32 ----
        v8f acc = {};
        const float* arow = &Alds[lm * AS + 2 * half];
        const float* brow = &Wlds[n  * WS + 2 * half];
        #pragma unroll
        for (int k0 = 0; k0 < KD; k0 += 4) {
            v2f a, b;
            a.x = arow[k0]; a.y = arow[k0 + 1];
            b.x = brow[k0]; b.y = brow[k0 + 1];
            acc = __builtin_amdgcn_wmma_f32_16x16x4_f32(
                false, a, false, b, (short)0, acc, false, false);
        }

        const float bv = blds[n];
        #pragma unroll
        for (int r = 0; r < 8; ++r) {
            float v = acc[r] + bv;
            if (relu) v = fmaxf(v, 0.0f);
            C[((size_t)t * 16 + r + 8 * half) * ND + n] = v;
        }
    }
}

// ---------------------------------------------------------------------------
// Generic WMMA fp32 GEMM (weights from global) for head layers with big K*N.
__global__ void __launch_bounds__(256)
gemm_wmma_kernel(const float* __restrict__ A, const float* __restrict__ W,
                 const float* __restrict__ bias, float* __restrict__ C,
                 int M, int K, int N, int relu) {
    const int lane = threadIdx.x & 31;
    const int wave = threadIdx.x >> 5;
    const int half = lane >> 4;
    const int lm   = lane & 15;
    const int m0   = blockIdx.x * 16;
    const int ntiles = N >> 4;

    const float* arow = A + (size_t)(m0 + lm) * K + 2 * half;

    for (int nt = wave; nt < ntiles; nt += 8) {
        const int n = (nt << 4) + lm;
        v8f acc = {};
        #pragma unroll 4
        for (int k0 = 0; k0 < K; k0 += 4) {
            v2f a, b;
            a.x = arow[k0];
            a.y = arow[k0 + 1];
            b.x = W[(size_t)(k0 + 2 * half)     * N + n];
            b.y = W[(size_t)(k0 + 2 * half + 1) * N + n];
            acc = __builtin_amdgcn_wmma_f32_16x16x4_f32(
                false, a, false, b, (short)0, acc, false, false);
        }
        const float bv = bias ? bias[n] : 0.0f;
        #pragma unroll
        for (int r = 0; r < 8; ++r) {
            float v = acc[r] + bv;
            if (relu) v = fmaxf(v, 0.0f);
            C[(size_t)(m0 + r + 8 * half) * N + n] = v;
        }
    }
}

// ---------------------------------------------------------------------------
__global__ void bn_stats_kernel(const float* __restrict__ h, float* __restrict__ stats, int nrows) {
    int gtid = blockIdx.x * blockDim.x + threadIdx.x;
    int c  = gtid & (DIMC - 1);
    int r0 = gtid >> 7;
    float s = 0.0f, s2 = 0.0f;
    for (int r = r0; r < nrows; r += 512) {
        float v = h[(size_t)r * DIMC + c];
        s  += v;
        s2 += v * v;
    }
    atomicAdd(&stats[c], s);
    atomicAdd(&stats[DIMC + c], s2);
}

__global__ void bn_apply_kernel(float* __restrict__ h, const float* __restrict__ stats,
                                const float* __restrict__ gamma, const float* __restrict__ beta,
                                int nrows) {
    size_t i = (size_t)blockIdx.x * blockDim.x + threadIdx.x;
    if (i >= (size_t)nrows * DIMC) return;
    int c = (int)(i & (DIMC - 1));
    float inv_n = 1.0f / (float)nrows;
    float mu  = stats[c] * inv_n;
    float var = stats[DIMC + c] * inv_n - mu * mu;
    h[i] = (h[i] - mu) * rsqrtf(var + BN_EPS) * gamma[c] + beta[c];
}

// ---------------------------------------------------------------------------
__global__ void pool_kernel(float* __restrict__ g, const float* __restrict__ h,
                            const int* __restrict__ batch, int nrows) {
    int gtid = blockIdx.x * blockDim.x + threadIdx.x;
    int n    = gtid >> 5;
    int lane = gtid & 31;
    if (n >= nrows) return;
    int b = batch[n];
    const float4* hs = (const float4*)(h + (size_t)n * DIMC);
    float4 v = hs[lane];
    float* gp = g + (size_t)b * DIMC + lane * 4;
    atomicAdd(gp + 0, v.x);
    atomicAdd(gp + 1, v.y);
    atomicAdd(gp + 2, v.z);
    atomicAdd(gp + 3, v.w);
}

// ---------------------------------------------------------------------------
__global__ void final_out_kernel(const float* __restrict__ r3, const float* __restrict__ W,
                                 const float* __restrict__ b, float* __restrict__ out) {
    __shared__ float red[256];
    int g = blockIdx.x, t = threadIdx.x;
    red[t] = r3[(size_t)g * 256 + t] * W[t];
    __syncthreads();
    for (int s = 128; s > 0; s >>= 1) {
        if (t < s) red[t] += red[t + s];
        __syncthreads();
    }
    if (t == 0) out[g] = red[0] + b[0];
}

// ---------------------------------------------------------------------------
extern "C" void kernel_launch(void* const* d_in, const int* in_sizes, int n_in,
                              void* d_out, int out_size, void* d_ws, size_t ws_size,
                              hipStream_t stream) {
    (void)in_sizes; (void)n_in; (void)out_size; (void)ws_size;

    const float* x     = (const float*)d_in[0];
    const int*   ei    = (const int*)d_in[1];       // [2,E]: row0 = src, row1 = dst
    const int*   batch = (const int*)d_in[2];
    const int*   src   = ei;
    const int*   dst   = ei + EE;

    float* P0    = (float*)d_ws;                    // agg
    float* P1    = P0 + (size_t)NN * DIMC;          // h1 (reused: pooled g)
    float* P2    = P1 + (size_t)NN * DIMC;          // h2 / layer output (in-place BN)
    float* stats = P2 + (size_t)NN * DIMC;          // 128 sums + 128 sumsqs

    const size_t nd      = (size_t)NN * DIMC;
    const int    nd_blks = (int)((nd + 255) / 256);              // 50000
    const int    sc_blks = (int)(((size_t)EE * 32 + 255) / 256); // 80000
    const int    pl_blks = (int)(((size_t)NN * 32 + 255) / 256); // 12500
    const int    gemm_blks = 1024;                               // grid-stride blocks

    const float* cur = x;
    for (int layer = 0; layer < 5; ++layer) {
        const int base = 3 + 6 * layer;
        const float* W1    = (const float*)d_in[base + 0];
        const float* b1    = (const float*)d_in[base + 1];
        const float* W2    = (const float*)d_in[base + 2];
        const float* b2    = (const float*)d_in[base + 3];
        const float* gamma = (const float*)d_in[base + 4];
        const float* beta  = (const float*)d_in[base + 5];

        copy_kernel<<<nd_blks, 256, 0, stream>>>(P0, cur, nd);
        edge_scatter_kernel<<<sc_blks, 256, 0, stream>>>(P0, cur, src, dst, EE);

        gemm_conv_kernel<DIMC, DIMC><<<gemm_blks, 256, 0, stream>>>(P0, W1, b1, P1, NN, 1);
        gemm_conv_kernel<DIMC, DIMC><<<gemm_blks, 256, 0, stream>>>(P1, W2, b2, P2, NN, 1);

        zero_kernel<<<1, 256, 0, stream>>>(stats, 256);
        bn_stats_kernel<<<256, 256, 0, stream>>>(P2, stats, NN);
        bn_apply_kernel<<<nd_blks, 256, 0, stream>>>(P2, stats, gamma, beta, NN);

        cur = P2;
    }

    // global add pool -> g[512,128]  (reuse P1)
    float* g = P1;
    zero_kernel<<<(NGRAPH * DIMC + 255) / 256, 256, 0, stream>>>(g, (size_t)NGRAPH * DIMC);
    pool_kernel<<<pl_blks, 256, 0, stream>>>(g, cur, batch, NN);

    const float* fc1xd_W = (const float*)d_in[33];
    const float* fc1xd_b = (const float*)d_in[34];
    const float* fc1_W   = (const float*)d_in[35];
    const float* fc1_b   = (const float*)d_in[36];
    const float* fc2_W   = (const float*)d_in[37];
    const float* fc2_b   = (const float*)d_in[38];
    const float* outW    = (const float*)d_in[39];
    const float* outB    = (const float*)d_in[40];

    float* r1 = P0;                                  // [512,128]
    float* r2 = r1 + (size_t)NGRAPH * 128;           // [512,1024]
    float* r3 = r2 + (size_t)NGRAPH * 1024;          // [512,256]

    gemm_conv_kernel<DIMC, DIMC><<<NGRAPH / 16, 256, 0, stream>>>(g, fc1xd_W, fc1xd_b, r1, NGRAPH, 1);
    gemm_wmma_kernel<<<NGRAPH / 16, 256, 0, stream>>>(r1, fc1_W, fc1_b, r2, NGRAPH, 128,  1024, 1);
    gemm_wmma_kernel<<<NGRAPH / 16, 256, 0, stream>>>(r2, fc2_W, fc2_b, r3, NGRAPH, 1024, 256,  1);
    final_out_kernel<<<NGRAPH, 256, 0, stream>>>(r3, outW, outB, (float*)d_out);
}